// LossComputation_22041772163705
// MI455X (gfx1250) — compile-verified
//
#include <hip/hip_runtime.h>
#include <hip/hip_bf16.h>
#include <math.h>

// ---------------- problem constants (from reference) ----------------
#define B_N      4096
#define D_N      1024
#define C_N      11003
#define SCALE_F  28.0f
#define EPS_LS_F 0.1f
#define T_F      4.0f
#define ALPHA_F  0.6f
#define BETA_F   0.4f
#define SPOS_F   10.0f
#define SNEG_F   40.0f
#define HDR_A    3.0f
#define PLEN_N   63
#define CT       688          // ceil(C_N/16); 688 % 8 == 0 -> uniform wave work
#define LDS_STRIDE 1026       // even stride => 8B-aligned ds_load_b64
// logits are SCALE * cos-sim, so bounded: static softmax shifts are safe
#define LMAX1    SCALE_F              // max |logit| at temperature 1
#define LMAXT    (SCALE_F / T_F)      // max |logit/T| at temperature T

typedef float v2f __attribute__((ext_vector_type(2)));
typedef float v8f __attribute__((ext_vector_type(8)));

// CDNA5 fp32 WMMA: D[16x16] = A[16x4] * B[4x16] + C  (wave32)
static __device__ __forceinline__ v8f wmma4(v2f a, v2f b, v8f c) {
  return __builtin_amdgcn_wmma_f32_16x16x4_f32(false, a, false, b, (short)0, c,
                                               false, false);
}

// numerically-stable log1p(exp(z))
static __device__ __forceinline__ float softplusf(float z) {
  return fmaxf(z, 0.0f) + log1pf(__expf(-fabsf(z)));
}

// ---------------- small helpers ----------------
__global__ void init_k(float* __restrict__ acc) {
  if (threadIdx.x < 8) acc[threadIdx.x] = 0.0f;
}

// one block per row: y = x / ||x||
__global__ void norm_rows_k(const float* __restrict__ x, float* __restrict__ y) {
  __shared__ float red[256];
  const int row = blockIdx.x;
  const float* xr = x + (size_t)row * D_N;
  float s = 0.0f;
  for (int i = threadIdx.x; i < D_N; i += 256) { float v = xr[i]; s += v * v; }
  red[threadIdx.x] = s;
  __syncthreads();
  for (int st = 128; st > 0; st >>= 1) {
    if (threadIdx.x < st) red[threadIdx.x] += red[threadIdx.x + st];
    __syncthreads();
  }
  const float inv = rsqrtf(red[0]);
  float* yr = y + (size_t)row * D_N;
  for (int i = threadIdx.x; i < D_N; i += 256) yr[i] = xr[i] * inv;
}

// column-normalize W [D,C] row-major and emit WMMA-packed layout:
// wp[((k>>2)*C + c)*4 + (k&3)] = W[k,c] / ||W[:,c]||
// => a lane's two K values for one column are one aligned 8B load.
__global__ void norm_cols_pack_k(const float* __restrict__ w, float* __restrict__ wp) {
  const int c = blockIdx.x * 256 + threadIdx.x;
  if (c >= C_N) return;
  float s = 0.0f;
  for (int k = 0; k < D_N; ++k) { float v = w[(size_t)k * C_N + c]; s += v * v; }
  const float inv = rsqrtf(s);
  for (int k = 0; k < D_N; ++k)
    wp[((size_t)(k >> 2) * C_N + c) * 4 + (k & 3)] = w[(size_t)k * C_N + c] * inv;
}

// ---------------- fused logits GEMM + per-row softmax statistics ------------
// Tile D = A(W cols as M, 16x4) x B(emb rows as N, 4x16).
// C-layout => lane owns row N=(lane&15) and cols M = col0 + (lane>>4)*8 + v.
// Logits bounded in [-SCALE, SCALE] -> branch-free static-shift softmax sums.
// stats[row*12]: lseA,sumA,pickA, lseB,sumB,pickB, lseC,sumC,pickC,
//                lseTA, lseTB, Srow(=sum softmax_T(a)*(a-b))
__global__ void logits_stats_k(const float* __restrict__ vn,
                               const float* __restrict__ tn,
                               const float* __restrict__ Wp,
                               const float* __restrict__ Wtp,
                               const int* __restrict__ labels,
                               float* __restrict__ stats) {
  extern __shared__ float smem[];
  float* sv   = smem;                         // 16*LDS_STRIDE
  float* stld = sv + 16 * LDS_STRIDE;         // 16*LDS_STRIDE
  float* sred = stld + 16 * LDS_STRIDE;       // 256*12
  const int tid = threadIdx.x;
  const int r0  = blockIdx.x * 16;

  for (int idx = tid; idx < 16 * D_N; idx += 256) {
    const int r = idx >> 10, c = idx & (D_N - 1);
    sv[r * LDS_STRIDE + c]   = vn[(size_t)(r0 + r) * D_N + c];
    stld[r * LDS_STRIDE + c] = tn[(size_t)(r0 + r) * D_N + c];
  }
  __syncthreads();

  const int lane = tid & 31, wave = tid >> 5;
  const int laneLo = lane & 15, hi = lane >> 4, klo = hi * 2;
  const int label = labels[r0 + laneLo];
  const float* svr = sv + laneLo * LDS_STRIDE;
  const float* str = stld + laneLo * LDS_STRIDE;

  float sA = 0.f, sumA = 0.f, pkA = 0.f;   // visual logits   (exp shifted by LMAX1)
  float sB = 0.f, sumB = 0.f, pkB = 0.f;   // textual logits
  float sC = 0.f, sumC = 0.f, pkC = 0.f;   // tn@Wtn logits
  float sTA = 0.f, wT = 0.f, sTB = 0.f;    // temp-T sums (shifted by LMAXT)

  for (int t = wave; t < CT; t += 8) {
    const int col0 = t * 16;
    // clamp instead of predicating loads: cols >= C_N are discarded in stats,
    // and clamping keeps EXEC uniform (required around WMMA) and branch-free.
    const int ca = min(col0 + laneLo, C_N - 1);
    v8f accA = {0.f, 0.f, 0.f, 0.f, 0.f, 0.f, 0.f, 0.f};
    v8f accB = {0.f, 0.f, 0.f, 0.f, 0.f, 0.f, 0.f, 0.f};
    v8f accC = {0.f, 0.f, 0.f, 0.f, 0.f, 0.f, 0.f, 0.f};
#pragma unroll 4
    for (int k = 0; k < D_N; k += 4) {
      const size_t o = ((size_t)(k >> 2) * C_N + ca) * 4 + klo;
      const v2f aW  = *(const v2f*)(Wp + o);    // aligned 8B: packed K pair
      const v2f aWt = *(const v2f*)(Wtp + o);
      const v2f bv  = *(const v2f*)(svr + k + klo);
      const v2f bt  = *(const v2f*)(str + k + klo);
      accA = wmma4(aW,  bv, accA);
      accB = wmma4(aW,  bt, accB);
      accC = wmma4(aWt, bt, accC);
    }
#pragma unroll
    for (int v = 0; v < 8; ++v) {
      const int col = col0 + hi * 8 + v;
      if (col < C_N) {
        const float a  = SCALE_F * accA[v];
        const float b  = SCALE_F * accB[v];
        const float c3 = SCALE_F * accC[v];
        sA += __expf(a - LMAX1);
        sB += __expf(b - LMAX1);
        sC += __expf(c3 - LMAX1);
        sumA += a; sumB += b; sumC += c3;
        if (col == label) { pkA = a; pkB = b; pkC = c3; }
        const float eTa = __expf(a * (1.0f / T_F) - LMAXT);
        sTA += eTa;
        wT  += eTa * (a - b);
        sTB += __expf(b * (1.0f / T_F) - LMAXT);
      }
    }
  }

  float* p = sred + tid * 12;
  p[0] = sA;  p[1] = sumA; p[2] = pkA;
  p[3] = sB;  p[4] = sumB; p[5] = pkB;
  p[6] = sC;  p[7] = sumC; p[8] = pkC;
  p[9] = sTA; p[10] = wT;  p[11] = sTB;
  __syncthreads();

  if (tid < 16) {      // one thread per row: sum the 16 lane-partials
    float g[12];
#pragma unroll
    for (int q = 0; q < 12; ++q) g[q] = 0.f;
    for (int w = 0; w < 8; ++w)
      for (int h = 0; h < 2; ++h) {
        const float* q = sred + (w * 32 + h * 16 + tid) * 12;
#pragma unroll
        for (int z = 0; z < 12; ++z) g[z] += q[z];
      }
    float* o = stats + (size_t)(r0 + tid) * 12;
    o[0] = LMAX1 + logf(g[0]);  o[1] = g[1];  o[2] = g[2];
    o[3] = LMAX1 + logf(g[3]);  o[4] = g[4];  o[5] = g[5];
    o[6] = LMAX1 + logf(g[6]);  o[7] = g[7];  o[8] = g[8];
    o[9]  = LMAXT + logf(g[9]);         // lseTA
    o[10] = LMAXT + logf(g[11]);        // lseTB
    o[11] = g[10] / g[9];               // Srow = wT / sTA (shift cancels)
  }
}

// ---------------- Gram matrix simV = vn @ vn.T via WMMA ----------------
__global__ void gram_k(const float* __restrict__ e, float* __restrict__ sim) {
  const int lane = threadIdx.x & 31;
  const int idx = blockIdx.x * 8 + (threadIdx.x >> 5);   // tile id
  const int i0 = (idx >> 8) * 16, j0 = (idx & 255) * 16;
  const int laneLo = lane & 15, hi = lane >> 4;
  const v2f* ar = (const v2f*)(e + (size_t)(i0 + laneLo) * D_N);
  const v2f* br = (const v2f*)(e + (size_t)(j0 + laneLo) * D_N);
  v8f acc = {0.f, 0.f, 0.f, 0.f, 0.f, 0.f, 0.f, 0.f};
#pragma unroll 4
  for (int k2 = 0; k2 < D_N / 2; k2 += 2) {   // k2 = k/2; lane pair at k/2 + hi
    acc = wmma4(ar[k2 + hi], br[k2 + hi], acc);
  }
#pragma unroll
  for (int v = 0; v < 8; ++v) {
    const int i = i0 + hi * 8 + v;
    sim[(size_t)i * B_N + j0 + laneLo] = acc[v];
  }
}

// ---------------- fused global-align: sim tile + softplus reduce ------------
__global__ void ga_k(const float* __restrict__ vn, const float* __restrict__ tn,
                     const int* __restrict__ labels, float* __restrict__ acc_out) {
  __shared__ float red[256];
  const int tid = threadIdx.x, lane = tid & 31;
  const int idx = blockIdx.x * 8 + (tid >> 5);
  const int i0 = (idx >> 8) * 16, j0 = (idx & 255) * 16;
  const int laneLo = lane & 15, hi = lane >> 4;
  const v2f* ar = (const v2f*)(vn + (size_t)(i0 + laneLo) * D_N);
  const v2f* br = (const v2f*)(tn + (size_t)(j0 + laneLo) * D_N);
  v8f acc = {0.f, 0.f, 0.f, 0.f, 0.f, 0.f, 0.f, 0.f};
#pragma unroll 4
  for (int k2 = 0; k2 < D_N / 2; k2 += 2) {
    acc = wmma4(ar[k2 + hi], br[k2 + hi], acc);
  }
  const int lj = labels[j0 + laneLo];
  float lsum = 0.f;
#pragma unroll
  for (int v = 0; v < 8; ++v) {
    const int i = i0 + hi * 8 + v;
    const float s = acc[v];
    lsum += (labels[i] == lj) ? softplusf(-SPOS_F * (s - ALPHA_F))
                              : softplusf( SNEG_F * (s - BETA_F));
  }
  red[tid] = lsum;
  __syncthreads();
  for (int st = 128; st > 0; st >>= 1) {
    if (tid < st) red[tid] += red[tid + st];
    __syncthreads();
  }
  if (tid == 0) atomicAdd(acc_out, red[0]);
}

// ---------------- feature L2 distillation ----------------
__global__ void distf_k(const float* __restrict__ vn, const float* __restrict__ tn,
                        float* __restrict__ acc_out) {
  __shared__ float red[256];
  float s = 0.f;
  const int n = B_N * D_N;
  for (int i = blockIdx.x * 256 + threadIdx.x; i < n; i += 256 * 256) {
    const float d = vn[i] - tn[i];
    s += d * d;
  }
  red[threadIdx.x] = s;
  __syncthreads();
  for (int st = 128; st > 0; st >>= 1) {
    if (threadIdx.x < st) red[threadIdx.x] += red[threadIdx.x + st];
    __syncthreads();
  }
  if (threadIdx.x == 0) atomicAdd(acc_out, red[0]);
}

// ---------------- HardDarkRank: one block per anchor row ----------------
__global__ void hdr_k(const float* __restrict__ simV, const float* __restrict__ tn,
                      float* __restrict__ acc_out) {
  __shared__ float row[B_N];
  __shared__ float tni[D_N];
  __shared__ float rv[256];
  __shared__ int   ri[256];
  __shared__ int   selj[PLEN_N];
  __shared__ float sels[PLEN_N];
  const int i = blockIdx.x, tid = threadIdx.x;
  for (int j = tid; j < B_N; j += 256) row[j] = simV[(size_t)i * B_N + j];
  for (int d = tid; d < D_N; d += 256) tni[d] = tn[(size_t)i * D_N + d];
  __syncthreads();
  if (tid == 0) row[i] = -2e30f;        // exclude self (teacher rank 0)
  __syncthreads();
  // top-63 by teacher similarity (score_t monotone in sim)
  for (int it = 0; it < PLEN_N; ++it) {
    float bv = -3e30f; int bi = 0;
    for (int j = tid; j < B_N; j += 256) {
      const float v = row[j];
      if (v > bv) { bv = v; bi = j; }
    }
    rv[tid] = bv; ri[tid] = bi;
    __syncthreads();
    for (int st = 128; st > 0; st >>= 1) {
      if (tid < st && rv[tid + st] > rv[tid]) { rv[tid] = rv[tid + st]; ri[tid] = ri[tid + st]; }
      __syncthreads();
    }
    if (tid == 0) { selj[it] = ri[0]; row[ri[0]] = -2e30f; }
    __syncthreads();
  }
  // student scores at the selected indices
  for (int r = 0; r < PLEN_N; ++r) {
    const float* tj = tn + (size_t)selj[r] * D_N;
    float p = 0.f;
    for (int d = tid; d < D_N; d += 256) p += tni[d] * tj[d];
    rv[tid] = p;
    __syncthreads();
    for (int st = 128; st > 0; st >>= 1) {
      if (tid < st) rv[tid] += rv[tid + st];
      __syncthreads();
    }
    if (tid == 0) {
      const float dd = sqrtf(fmaxf(2.f - 2.f * rv[0], 1e-12f));
      sels[r] = -HDR_A * dd * dd * dd;
    }
    __syncthreads();
  }
  if (tid == 0) {   // suffix logsumexp, listwise -log prob
    float tail = sels[PLEN_N - 1];
    float lp = 0.f;
    for (int r = PLEN_N - 2; r >= 0; --r) {
      const float o = sels[r];
      const float mx = fmaxf(o, tail);
      tail = mx + log1pf(__expf(fminf(o, tail) - mx));
      lp += o - tail;
    }
    atomicAdd(acc_out, -lp);
  }
}

// ---------------- final combine ----------------
__global__ void final_k(const float* __restrict__ stats, const float* __restrict__ acc,
                        float* __restrict__ out) {
  __shared__ float red[4][256];
  float vsum = 0.f, tsum = 0.f, fsum = 0.f, klsum = 0.f;
  for (int r = threadIdx.x; r < B_N; r += 256) {
    const float* st = stats + (size_t)r * 12;
    const float lseA = st[0], sumA = st[1], pkA = st[2];
    const float lseB = st[3], sumB = st[4], pkB = st[5];
    const float lseC = st[6], sumC = st[7], pkC = st[8];
    const float lseTA = st[9], lseTB = st[10], Sr = st[11];
    vsum  += -((1.f - EPS_LS_F) * (pkA - lseA) + (EPS_LS_F / C_N) * (sumA - (float)C_N * lseA));
    tsum  += -((1.f - EPS_LS_F) * (pkB - lseB) + (EPS_LS_F / C_N) * (sumB - (float)C_N * lseB));
    fsum  += -((1.f - EPS_LS_F) * (pkC - lseC) + (EPS_LS_F / C_N) * (sumC - (float)C_N * lseC));
    klsum += Sr * (1.f / T_F) - lseTA + lseTB;
  }
  red[0][threadIdx.x] = vsum; red[1][threadIdx.x] = tsum;
  red[2][threadIdx.x] = fsum; red[3][threadIdx.x] = klsum;
  __syncthreads();
  for (int st = 128; st > 0; st >>= 1) {
    if (threadIdx.x < st)
      for (int q = 0; q < 4; ++q) red[q][threadIdx.x] += red[q][threadIdx.x + st];
    __syncthreads();
  }
  if (threadIdx.x == 0) {
    const float invB = 1.f / (float)B_N;
    const float instance = (red[0][0] + red[1][0] + red[2][0] + red[3][0]) * invB
                         + acc[1] * invB            // dist_f
                         + 0.1f * acc[2] * invB;    // HardDarkRank
    out[0] = instance + 2.f * acc[0] * invB;        // + global_align
  }
}

// ---------------- host launch ----------------
extern "C" void kernel_launch(void* const* d_in, const int* in_sizes, int n_in,
                              void* d_out, int out_size, void* d_ws, size_t ws_size,
                              hipStream_t stream) {
  const float* visual  = (const float*)d_in[0];
  const float* textual = (const float*)d_in[1];
  const int*   labels  = (const int*)d_in[2];
  const float* W       = (const float*)d_in[3];
  const float* W_text  = (const float*)d_in[4];
  float* out = (float*)d_out;

  float* ws = (float*)d_ws;
  float* vn    = ws;
  float* tn    = vn  + (size_t)B_N * D_N;
  float* Wp    = tn  + (size_t)B_N * D_N;     // packed normalized W
  float* Wtp   = Wp  + (size_t)D_N * C_N;     // packed normalized W_text
  float* simV  = Wtp + (size_t)D_N * C_N;
  float* stats = simV + (size_t)B_N * B_N;
  float* acc   = stats + (size_t)B_N * 12;

  init_k<<<1, 32, 0, stream>>>(acc);
  norm_rows_k<<<B_N, 256, 0, stream>>>(visual, vn);
  norm_rows_k<<<B_N, 256, 0, stream>>>(textual, tn);
  norm_cols_pack_k<<<(C_N + 255) / 256, 256, 0, stream>>>(W, Wp);
  norm_cols_pack_k<<<(C_N + 255) / 256, 256, 0, stream>>>(W_text, Wtp);

  const size_t shmem = (size_t)(2 * 16 * LDS_STRIDE + 256 * 12) * sizeof(float);
  logits_stats_k<<<B_N / 16, 256, shmem, stream>>>(vn, tn, Wp, Wtp, labels, stats);

  gram_k<<<(B_N / 16) * (B_N / 16) / 8, 256, 0, stream>>>(vn, simV);
  ga_k<<<(B_N / 16) * (B_N / 16) / 8, 256, 0, stream>>>(vn, tn, labels, &acc[0]);
  distf_k<<<256, 256, 0, stream>>>(vn, tn, &acc[1]);
  hdr_k<<<B_N, 256, 0, stream>>>(simV, tn, &acc[2]);
  final_k<<<1, 256, 0, stream>>>(stats, acc, out);
}